// SimplifiedRelativePositionalEmbedding_37855841747306
// MI455X (gfx1250) — compile-verified
//
#include <hip/hip_runtime.h>

typedef float v2f __attribute__((ext_vector_type(2)));
typedef float v8f __attribute__((ext_vector_type(8)));
typedef unsigned int v4u __attribute__((ext_vector_type(4)));
typedef int v4i __attribute__((ext_vector_type(4)));
typedef int v8i __attribute__((ext_vector_type(8)));

#define T_ 2048
#define D_ 512
#define H_ 8
#define TP1 2049
#define ROWS 16          // output rows per tile
#define LROWS 31         // LDS rows: j = i0-15 .. i0+15
#define LSTRIDE 520      // floats per LDS row (512 + 2x4 pad -> conflict-free)
#define OUTSTRIDE (9 * 512)

// TDM path: available when the tensor-DMA builtin exists (device pass only).
#if defined(__AMDGCN__) && __has_builtin(__builtin_amdgcn_tensor_load_to_lds) && \
    __has_builtin(__builtin_amdgcn_s_wait_tensorcnt)
#define USE_TDM 1
#else
#define USE_TDM 0
#endif

// LDS element (r, c): pad_interval=256 dwords / pad_amount=4 dwords inserts a
// 4-float gap after each 256 floats -> offset = r*520 + c + (c>=256 ? 4 : 0).
__device__ __forceinline__ int lds_off(int r, int c) {
    return r * LSTRIDE + c + ((c >= 256) ? 4 : 0);
}

__global__ __launch_bounds__(256) void srpe_wmma_kernel(const float* __restrict__ X,
                                                        const float* __restrict__ W,
                                                        float* __restrict__ out) {
    __shared__ float lds[LROWS * LSTRIDE];
    const int i0  = blockIdx.x * ROWS;
    const int b   = blockIdx.y;
    const int tid = threadIdx.x;

    const float4* X4 = (const float4*)X;

    // valid source-row range for this tile (rows outside get exact zeros;
    // their attention weights are identically zero for all stored rows)
    const int rows_lo = (i0 - 15 < 0) ? 0 : (i0 - 15);
    const int rows_hi = (i0 + 15 > T_ - 1) ? (T_ - 1) : (i0 + 15);
    const int nrows   = rows_hi - rows_lo + 1;
    const int skip    = rows_lo - (i0 - 15);   // leading zero rows in LDS

#if USE_TDM
    // ---- zero-fill out-of-range LDS rows (edge blocks only do real work) ----
    for (int idx = tid; idx < LROWS * (D_ / 4); idx += 256) {
        int r  = idx >> 7;          // / 128
        int c4 = idx & 127;
        int j  = i0 - 15 + r;
        if (j < 0 || j >= T_) {
            float4 z; z.x = z.y = z.z = z.w = 0.0f;
            *(float4*)&lds[lds_off(r, 4 * c4)] = z;
        }
    }
    // ---- wave 0 issues one 2D TDM transfer: nrows x 512 fp32, LDS-padded ----
    if (tid < 32) {
        unsigned long long ga =
            (unsigned long long)(const void*)(X + (size_t)(b * T_ + rows_lo) * D_);
        unsigned int la = (unsigned int)(size_t)&lds[skip * LSTRIDE];
        v4u g0;
        g0[0] = 1u;                                   // count=1 (valid), user mode
        g0[1] = la;                                   // lds_addr (bytes)
        g0[2] = (unsigned int)ga;                     // global_addr[31:0]
        g0[3] = ((unsigned int)(ga >> 32) & 0x01FFFFFFu) | 0x80000000u; // addr hi | type=2
        v8i g1;
        g1[0] = (int)0x07D20000;    // data_size=4B, pad_en, interval=256dw, amount=4dw
        g1[1] = (int)(512u << 16);  // tensor_dim0 = 512 (bits 63:48)
        g1[2] = (int)((unsigned)nrows << 16); // tensor_dim1 lo (bits 95:80)
        g1[3] = (int)(512u << 16);  // tile_dim0 = 512 (bits 127:112)
        g1[4] = nrows;              // tile_dim1 (bits 143:128)
        g1[5] = 512;                // tensor_dim0_stride (bits 191:160)
        g1[6] = 0;
        g1[7] = 0;
        v4i z4; z4[0] = z4[1] = z4[2] = z4[3] = 0;    // groups 2/3 unused (2D)
#if __clang_major__ >= 23
        v8i z8; z8[0]=z8[1]=z8[2]=z8[3]=z8[4]=z8[5]=z8[6]=z8[7]=0;
        __builtin_amdgcn_tensor_load_to_lds(g0, g1, z4, z4, z8, 0);
#else
        __builtin_amdgcn_tensor_load_to_lds(g0, g1, z4, z4, 0);
#endif
        __builtin_amdgcn_s_wait_tensorcnt(0);
    }
#else
    // ---- fallback cooperative fill (zeros outside [0,T)) ----
    for (int idx = tid; idx < LROWS * (D_ / 4); idx += 256) {
        int r  = idx >> 7;
        int c4 = idx & 127;
        int j  = i0 - 15 + r;
        float4 val; val.x = val.y = val.z = val.w = 0.0f;
        if (j >= 0 && j < T_) val = X4[(size_t)(b * T_ + j) * (D_ / 4) + c4];
        *(float4*)&lds[lds_off(r, 4 * c4)] = val;
    }
#endif

    // ---- X_tilde passthrough: out[b, i, 0:512] (row T is zeros) ----
    for (int idx = tid; idx < ROWS * (D_ / 4); idx += 256) {
        int m  = idx >> 7;
        int c4 = idx & 127;
        int i  = i0 + m;
        if (i <= T_) {
            float4 val;
            if (i < T_) {
                val = X4[(size_t)(b * T_ + i) * (D_ / 4) + c4];
            } else {
                val.x = val.y = val.z = val.w = 0.0f;
            }
            *(float4*)&out[((size_t)b * TP1 + i) * OUTSTRIDE + 4 * c4] = val;
        }
    }
    __syncthreads();

    const int h    = tid >> 5;      // wave id == head
    const int lane = tid & 31;
    const int m    = lane & 15;     // A row / B,C,D column
    const int half = lane >> 4;     // lane-half: K/M split
    const int i    = i0 + m;        // output row for A weights

    // ---- softmax weights for (i, h): deltas h..h+7, valid iff delta <= i-1 ----
    float wv[8];
#pragma unroll
    for (int t = 0; t < 8; ++t) wv[t] = W[(h + t) * H_ + h];
    float mx = -3.0e38f;
#pragma unroll
    for (int t = 0; t < 8; ++t) mx = (i >= h + t + 1) ? fmaxf(mx, wv[t]) : mx;
    mx = (i >= h + 1) ? mx : 0.0f;  // fully-masked row -> weights stay 0
    float wgt[8];
    float sum = 0.0f;
#pragma unroll
    for (int t = 0; t < 8; ++t) {
        float e = (i >= h + t + 1) ? expf(wv[t] - mx) : 0.0f;
        wgt[t] = e;
        sum += e;
    }
    float rs = 1.0f / fmaxf(sum, 1e-30f);
#pragma unroll
    for (int t = 0; t < 8; ++t) wgt[t] *= rs;

    // ---- A fragments (16x24 band matrix as 6 chunks of 16x4 fp32) ----
    // A[m][k] nonzero iff t = m+7-k in [0,7]; then weight = wgt[t] (delta = h+t)
    v2f afrag[6];
#pragma unroll
    for (int kc = 0; kc < 6; ++kc) {
#pragma unroll
        for (int v = 0; v < 2; ++v) {
            int k = 4 * kc + 2 * half + v;
            int t = m + 7 - k;
            afrag[kc][v] = (t >= 0 && t <= 7) ? wgt[t] : 0.0f;
        }
    }

    // LDS row for global k: r = k + 7 - h  (range [0,30] for h in [0,7])
    const int rbase = 7 - h;

    float inv[8];
#pragma unroll
    for (int r = 0; r < 8; ++r) inv[r] = 0.0f;

    // ---- pass A: accumulate per-row sum of squares over all 32 column tiles ----
    for (int ct = 0; ct < D_ / 16; ++ct) {
        int col = ct * 16 + m;
        v8f c = {};
#pragma unroll
        for (int kc = 0; kc < 6; ++kc) {
            int k0 = 4 * kc + 2 * half;
            v2f bfrag;
            bfrag[0] = lds[lds_off(rbase + k0,     col)];
            bfrag[1] = lds[lds_off(rbase + k0 + 1, col)];
            c = __builtin_amdgcn_wmma_f32_16x16x4_f32(false, afrag[kc], false, bfrag,
                                                      (short)0, c, false, false);
        }
#pragma unroll
        for (int r = 0; r < 8; ++r) inv[r] += c[r] * c[r];
    }
    // reduce across the 16 lanes of each half (row m lives in one lane-half)
#pragma unroll
    for (int r = 0; r < 8; ++r) {
        float s = inv[r];
        s += __shfl_xor(s, 1, 32);
        s += __shfl_xor(s, 2, 32);
        s += __shfl_xor(s, 4, 32);
        s += __shfl_xor(s, 8, 32);
        inv[r] = 1.0f / (sqrtf(s) + 1e-7f);
    }

    // ---- pass B: recompute (compute is free; memory-bound) and store scaled ----
    const size_t obase = ((size_t)b * TP1) * OUTSTRIDE + (size_t)(1 + h) * D_;
    for (int ct = 0; ct < D_ / 16; ++ct) {
        int col = ct * 16 + m;
        v8f c = {};
#pragma unroll
        for (int kc = 0; kc < 6; ++kc) {
            int k0 = 4 * kc + 2 * half;
            v2f bfrag;
            bfrag[0] = lds[lds_off(rbase + k0,     col)];
            bfrag[1] = lds[lds_off(rbase + k0 + 1, col)];
            c = __builtin_amdgcn_wmma_f32_16x16x4_f32(false, afrag[kc], false, bfrag,
                                                      (short)0, c, false, false);
        }
#pragma unroll
        for (int r = 0; r < 8; ++r) {
            int row = i0 + 8 * half + r;
            if (row <= T_) {
                out[obase + (size_t)row * OUTSTRIDE + (size_t)(ct * 16 + m)] = c[r] * inv[r];
            }
        }
    }
}

extern "C" void kernel_launch(void* const* d_in, const int* in_sizes, int n_in,
                              void* d_out, int out_size, void* d_ws, size_t ws_size,
                              hipStream_t stream) {
    const float* X = (const float*)d_in[0];   // (B, T, D) fp32
    const float* W = (const float*)d_in[1];   // (T, H)    fp32
    float* out = (float*)d_out;               // (B, T+1, (H+1)*D) fp32

    int B = in_sizes[0] / (T_ * D_);
    dim3 grid((TP1 + ROWS - 1) / ROWS, B);    // 129 x B workgroups
    srpe_wmma_kernel<<<grid, 256, 0, stream>>>(X, W, out);
}